// E8PQuantizedWeights_29317446762951
// MI455X (gfx1250) — compile-verified
//
#include <hip/hip_runtime.h>
#include <hip/hip_bf16.h>

// E8P codebook dequantization for MI455X (gfx1250).
// out[o, i] = grid[weight_q[o, i/8]][i%8] * scale
// Pure HBM-bandwidth-bound gather: ~200 MB moved -> ~8.7 us floor at 23.3 TB/s.
// CDNA5 path: double-buffered GLOBAL_LOAD_ASYNC_TO_LDS_B128 staging of the code
// stream + s_wait_asynccnt, cached gathers from the 2 MB codebook (L2-resident),
// and non-temporal 128-bit output stores to avoid polluting L2.

typedef __attribute__((ext_vector_type(4))) float v4f;
typedef __attribute__((ext_vector_type(4))) int   v4i;

#define OUT_FEATURES 4096
#define IN_FEATURES  11008
#define CODESZ       8

#define TPB   256          // threads per block (8 wave32s)
#define CPT   4            // codes per thread per chunk (16 B async load)
#define CHUNK (TPB * CPT)  // 1024 codes per block-chunk

#if defined(__AMDGCN__) && defined(__gfx1250__) && \
    __has_builtin(__builtin_amdgcn_global_load_async_to_lds_b128) && \
    __has_builtin(__builtin_amdgcn_s_wait_asynccnt)
#define USE_ASYNC_LDS 1
#else
#define USE_ASYNC_LDS 0
#endif

#if USE_ASYNC_LDS
// The builtin's params are v4i pointers in addrspace(1) (global) / addrspace(3) (LDS).
typedef v4i __attribute__((address_space(1))) gv4i;
typedef v4i __attribute__((address_space(3))) sv4i;

__device__ __forceinline__ void async_copy16(const int* gsrc, int* lds_dst) {
    __builtin_amdgcn_global_load_async_to_lds_b128(
        (gv4i*)gsrc, (sv4i*)lds_dst, /*offset=*/0, /*cpol=*/0);
}
#endif

__global__ __launch_bounds__(TPB) void e8p_dequant_kernel(
    const int*   __restrict__ wq,      // (O, I/8) codes, 0..65535
    const float* __restrict__ grid,    // (65536, 8) codebook
    const float* __restrict__ scale_p, // (1,)
    float*       __restrict__ out,     // (O, I)
    long long n_codes)
{
    const float s   = scale_p[0];          // uniform -> s_load_b32
    const int   tid = (int)threadIdx.x;

    const long long nchunks = n_codes / CHUNK;
    const long long cstride = (long long)gridDim.x;

#if USE_ASYNC_LDS
    __shared__ int codes[2][CHUNK];

    // Prologue: kick off the async DMA for this block's first chunk.
    long long c0 = (long long)blockIdx.x;
    if (c0 < nchunks) {
        async_copy16(wq + c0 * CHUNK + (long long)tid * CPT,
                     &codes[0][tid * CPT]);
    }

    int buf = 0;
    for (long long c = c0; c < nchunks; c += cstride) {
        const long long cn = c + cstride;
        if (cn < nchunks) {
            // Issue next chunk's DMA into the other buffer, then wait until
            // only that one is outstanding (async ops retire in order).
            async_copy16(wq + cn * CHUNK + (long long)tid * CPT,
                         &codes[buf ^ 1][tid * CPT]);
            __builtin_amdgcn_s_wait_asynccnt(1);
        } else {
            __builtin_amdgcn_s_wait_asynccnt(0);
        }

        // Each lane reads back exactly the 4 codes its own DMA wrote.
        v4i c4 = *(const v4i*)&codes[buf][tid * CPT];

        const long long base = (c * CHUNK + (long long)tid * CPT);
#pragma unroll
        for (int k = 0; k < CPT; ++k) {
            const unsigned code = (unsigned)c4[k] & 0xFFFFu;
            const v4f* g = (const v4f*)(grid + (long long)code * CODESZ);
            v4f lo = g[0] * s;                       // cached (RT) gathers
            v4f hi = g[1] * s;
            v4f* o = (v4f*)(out + (base + k) * CODESZ);
            __builtin_nontemporal_store(lo, o);      // streaming NT stores
            __builtin_nontemporal_store(hi, o + 1);
        }
        buf ^= 1;
    }
#else
    // Fallback: direct non-temporal 128-bit code loads.
    for (long long c = (long long)blockIdx.x; c < nchunks; c += cstride) {
        const long long base = c * CHUNK + (long long)tid * CPT;
        v4i c4 = __builtin_nontemporal_load((const v4i*)(wq + base));
#pragma unroll
        for (int k = 0; k < CPT; ++k) {
            const unsigned code = (unsigned)c4[k] & 0xFFFFu;
            const v4f* g = (const v4f*)(grid + (long long)code * CODESZ);
            v4f lo = g[0] * s;
            v4f hi = g[1] * s;
            v4f* o = (v4f*)(out + (base + k) * CODESZ);
            __builtin_nontemporal_store(lo, o);
            __builtin_nontemporal_store(hi, o + 1);
        }
    }
#endif

    // Tail (n_codes % CHUNK) — empty for the reference shapes, kept for safety.
    const long long tail_start = nchunks * CHUNK;
    for (long long j = tail_start + (long long)blockIdx.x * TPB + tid;
         j < n_codes;
         j += (long long)gridDim.x * TPB) {
        const unsigned code = (unsigned)wq[j] & 0xFFFFu;
        const v4f* g = (const v4f*)(grid + (long long)code * CODESZ);
        v4f lo = g[0] * s;
        v4f hi = g[1] * s;
        v4f* o = (v4f*)(out + j * CODESZ);
        __builtin_nontemporal_store(lo, o);
        __builtin_nontemporal_store(hi, o + 1);
    }
}

extern "C" void kernel_launch(void* const* d_in, const int* in_sizes, int n_in,
                              void* d_out, int out_size, void* d_ws, size_t ws_size,
                              hipStream_t stream) {
    const int*   wq    = (const int*)d_in[0];    // (4096, 1376) int32
    const float* grid  = (const float*)d_in[1];  // (65536, 8) f32
    const float* scale = (const float*)d_in[2];  // (1,) f32
    float*       out   = (float*)d_out;          // (4096, 11008) f32

    const long long n_codes = (long long)in_sizes[0];   // 5,636,096
    long long nchunks = n_codes / CHUNK;                // 5,504

    // ~4 pipelined chunks per block; plenty of waves to fill the WGP array.
    long long blocks = 1376;
    if (blocks > nchunks && nchunks > 0) blocks = nchunks;
    if (blocks < 1) blocks = 1;

    e8p_dequant_kernel<<<dim3((unsigned)blocks), dim3(TPB), 0, stream>>>(
        wq, grid, scale, out, n_codes);
}